// GNNModel_43860206027181
// MI455X (gfx1250) — compile-verified
//
#include <hip/hip_runtime.h>

// GCN (8 layers, D=128) on 50000 nodes / 600000 edges, f32 throughout.
// GEMM: V_WMMA_F32_16X16X4_F32 (full f32 precision on the matrix pipes),
// A-strip staged in LDS (shared by all 8 waves of the block) via
// GLOBAL_LOAD_ASYNC_TO_LDS_B128 (ASYNCcnt path). Scatter: hardware FP32
// global atomics. Whole working set (~80 MB) lives in MI455X's 192 MB L2.

#define N_NODES 50000
#define D 128
#define N_LAYERS 8
#define NEG_SLOPE 0.1f
#define A_PITCH 132   // floats; 528B row pitch: 16B aligned, bank-conflict-free

typedef float v2f __attribute__((ext_vector_type(2)));
typedef float v8f __attribute__((ext_vector_type(8)));
typedef int   v4i __attribute__((ext_vector_type(4)));

// builtin signature (from hipcc diagnostic): first param is AS(1) v4i*
typedef __attribute__((address_space(1))) v4i* gv4i_p;
typedef __attribute__((address_space(3))) v4i* lv4i_p;

#if defined(__has_builtin)
#if __has_builtin(__builtin_amdgcn_global_load_async_to_lds_b128) && \
    __has_builtin(__builtin_amdgcn_s_wait_asynccnt)
#define USE_ASYNC_LDS 1
#endif
#endif

// ---------------- degree / normalization ----------------

__global__ void deg_init(float* deg, int n) {
    int i = blockIdx.x * blockDim.x + threadIdx.x;
    if (i < n) deg[i] = 1.0f;  // self-loop
}

__global__ void deg_count(const int* __restrict__ dst, float* deg, int n_edges) {
    int e = blockIdx.x * blockDim.x + threadIdx.x;
    if (e < n_edges) unsafeAtomicAdd(&deg[dst[e]], 1.0f);
}

__global__ void deg_to_dinv(float* deg, int n) {
    int i = blockIdx.x * blockDim.x + threadIdx.x;
    if (i < n) {
        float d = deg[i];
        deg[i] = d > 0.0f ? rsqrtf(d) : 0.0f;
    }
}

// ---------------- dense GEMM: tmp = H (n x 128) @ W (128 x 128) ----------------
// One 256-thread block (8 waves) per 16-row strip; wave w owns columns
// [16w, 16w+16). 50000 % 16 == 0, so no predication: EXEC stays all-ones for
// WMMA. A layout (16x4 f32): lane<16 -> K=0,1 ; lane>=16 -> K=2,3, M = lane&15.
// B layout (4x16 f32): same halves over K, N = lane&15.

__global__ void __launch_bounds__(256)
gemm_h_w(const float* __restrict__ H, const float* __restrict__ W,
         float* __restrict__ out) {
    __shared__ float As[16 * A_PITCH];

    const int tid  = threadIdx.x;
    const int wave = tid >> 5;
    const int lane = tid & 31;
    const int half = lane >> 4;   // 0 or 1
    const int l15  = lane & 15;

    const int m0 = blockIdx.x * 16;
    const int n0 = wave * 16;

    // ---- stage A strip (16 x 128 f32 = 8 KB) into LDS, float4 granularity ----
    {
        int idx = tid;                       // 512 float4 chunks, 2 per thread
        #pragma unroll
        for (int r = 0; r < 2; ++r, idx += 256) {
            const int row = idx >> 5;        // 0..15
            const int f4  = idx & 31;        // 0..31
            const float* gsrc = H + (size_t)(m0 + row) * D + f4 * 4;
            float* ldst = &As[row * A_PITCH + f4 * 4];
#if defined(USE_ASYNC_LDS)
            __builtin_amdgcn_global_load_async_to_lds_b128(
                (gv4i_p)gsrc, (lv4i_p)ldst, 0, 0);
#else
            *(float4*)ldst = *(const float4*)gsrc;
#endif
        }
    }
#if defined(USE_ASYNC_LDS)
    __builtin_amdgcn_s_wait_asynccnt(0);
#endif
    __syncthreads();

    const float* __restrict__ bcol = W + n0 + l15;
    const float* __restrict__ arow = &As[l15 * A_PITCH];

    v8f acc = {};
    for (int k = 0; k < D; k += 4) {
        const int ka = k + half * 2;
        v2f a, b;
        a.x = arow[ka];                       // ds_load (conflict-free pitch)
        a.y = arow[ka + 1];
        b.x = bcol[(size_t)ka * D];
        b.y = bcol[(size_t)(ka + 1) * D];
        // (neg_a, A, neg_b, B, c_mod, C, reuse_a, reuse_b)
        acc = __builtin_amdgcn_wmma_f32_16x16x4_f32(
            false, a, false, b, (short)0, acc, false, false);
    }

    // C/D layout: VGPR v holds row m0 + v + 8*half, col n0 + l15.
    float* __restrict__ o = out + (size_t)(m0 + half * 8) * D + n0 + l15;
    #pragma unroll
    for (int v = 0; v < 8; ++v) o[(size_t)v * D] = acc[v];
}

// ---------------- aggregation ----------------

// agg[i][:] = dinv[i]^2 * tmp[i][:]   (self-loop term; also zero-initializes)
__global__ void init_agg(const float* __restrict__ tmp,
                         const float* __restrict__ dinv,
                         float* __restrict__ agg) {
    int idx = blockIdx.x * blockDim.x + threadIdx.x;   // over N*D/4
    if (idx >= N_NODES * (D / 4)) return;
    int node = idx / (D / 4);
    int f4   = idx % (D / 4);
    float di = dinv[node];
    float s  = di * di;
    float4 v = ((const float4*)(tmp + (size_t)node * D))[f4];
    float4 r = make_float4(s * v.x, s * v.y, s * v.z, s * v.w);
    ((float4*)(agg + (size_t)node * D))[f4] = r;
}

// agg[dst][:] += dinv[src]*dinv[dst] * tmp[src][:]
// 32 lanes per edge, each lane owns a float4 feature chunk.
__global__ void edge_scatter(const int* __restrict__ src,
                             const int* __restrict__ dst,
                             const float* __restrict__ dinv,
                             const float* __restrict__ tmp,
                             float* __restrict__ agg, int n_edges) {
    long long idx = (long long)blockIdx.x * blockDim.x + threadIdx.x;
    int e = (int)(idx >> 5);
    if (e >= n_edges) return;
    int g = (int)(idx & 31);
    int s = src[e], d = dst[e];
    float norm = dinv[s] * dinv[d];
    float4 v = ((const float4*)(tmp + (size_t)s * D))[g];
    float* o = agg + (size_t)d * D + g * 4;
    unsafeAtomicAdd(o + 0, norm * v.x);
    unsafeAtomicAdd(o + 1, norm * v.y);
    unsafeAtomicAdd(o + 2, norm * v.z);
    unsafeAtomicAdd(o + 3, norm * v.w);
}

// in-place: agg = leaky_relu(agg + b)
__global__ void bias_relu(float* __restrict__ agg, const float* __restrict__ b) {
    int idx = blockIdx.x * blockDim.x + threadIdx.x;   // over N*D
    if (idx >= N_NODES * D) return;
    float v = agg[idx] + b[idx & (D - 1)];
    agg[idx] = v > 0.0f ? v : NEG_SLOPE * v;
}

// ---------------- output head: 128 -> 1 ----------------

// one wave per node: tmp1[i] = dot(h[i], Wout)
__global__ void gemv_out(const float* __restrict__ h,
                         const float* __restrict__ Wout,
                         float* __restrict__ tmp1, int n) {
    int gtid = blockIdx.x * blockDim.x + threadIdx.x;
    int node = gtid >> 5;
    int lane = threadIdx.x & 31;
    if (node >= n) return;
    float4 hv = ((const float4*)(h + (size_t)node * D))[lane];
    float4 wv = ((const float4*)Wout)[lane];
    float p = hv.x * wv.x + hv.y * wv.y + hv.z * wv.z + hv.w * wv.w;
    #pragma unroll
    for (int off = 16; off > 0; off >>= 1) p += __shfl_down(p, off, 32);
    if (lane == 0) tmp1[node] = p;
}

__global__ void out_init(const float* __restrict__ tmp1,
                         const float* __restrict__ dinv,
                         float* __restrict__ aggs, int n) {
    int i = blockIdx.x * blockDim.x + threadIdx.x;
    if (i < n) aggs[i] = dinv[i] * dinv[i] * tmp1[i];
}

__global__ void out_scatter(const int* __restrict__ src,
                            const int* __restrict__ dst,
                            const float* __restrict__ dinv,
                            const float* __restrict__ tmp1,
                            float* __restrict__ aggs, int n_edges) {
    int e = blockIdx.x * blockDim.x + threadIdx.x;
    if (e < n_edges) {
        int s = src[e], d = dst[e];
        unsafeAtomicAdd(&aggs[d], dinv[s] * dinv[d] * tmp1[s]);
    }
}

__global__ void out_finish(const float* __restrict__ aggs,
                           const float* __restrict__ bout,
                           float* __restrict__ out, int n) {
    int i = blockIdx.x * blockDim.x + threadIdx.x;
    if (i < n) out[i] = aggs[i] + bout[0];
}

// ---------------- host orchestration ----------------

extern "C" void kernel_launch(void* const* d_in, const int* in_sizes, int n_in,
                              void* d_out, int out_size, void* d_ws, size_t ws_size,
                              hipStream_t stream) {
    const float* x    = (const float*)d_in[0];   // [N, 128]
    const float* Ws   = (const float*)d_in[1];   // [8, 128, 128]
    const float* bs   = (const float*)d_in[2];   // [8, 128]
    const float* Wout = (const float*)d_in[3];   // [128, 1]
    const float* bout = (const float*)d_in[4];   // [1]
    const int*   ei   = (const int*)d_in[5];     // [2, E]
    const int n_edges = in_sizes[5] / 2;
    const int* src = ei;
    const int* dst = ei + n_edges;

    // workspace layout (floats): dinv[N] | tmp[N*D] | aggA[N*D] | aggB[N*D]
    float* ws   = (float*)d_ws;
    float* dinv = ws;
    float* tmp  = ws + N_NODES;
    float* aggA = tmp + (size_t)N_NODES * D;
    float* aggB = aggA + (size_t)N_NODES * D;

    const int T = 256;

    // degree -> dinv (deg buffer reused in place)
    deg_init<<<(N_NODES + T - 1) / T, T, 0, stream>>>(dinv, N_NODES);
    deg_count<<<(n_edges + T - 1) / T, T, 0, stream>>>(dst, dinv, n_edges);
    deg_to_dinv<<<(N_NODES + T - 1) / T, T, 0, stream>>>(dinv, N_NODES);

    const float* h = x;
    float* bufs[2] = {aggA, aggB};
    const int gemm_blocks = N_NODES / 16;                       // 3125, exact
    const int agg4_blocks = (N_NODES * (D / 4) + T - 1) / T;
    const long long sc_threads = (long long)n_edges * 32;
    const unsigned sc_blocks = (unsigned)((sc_threads + T - 1) / T);
    const int nd_blocks = (N_NODES * D + T - 1) / T;

    for (int l = 0; l < N_LAYERS; ++l) {
        float* agg = bufs[l & 1];
        gemm_h_w<<<gemm_blocks, T, 0, stream>>>(h, Ws + (size_t)l * D * D, tmp);
        init_agg<<<agg4_blocks, T, 0, stream>>>(tmp, dinv, agg);
        edge_scatter<<<sc_blocks, T, 0, stream>>>(src, dst, dinv, tmp, agg, n_edges);
        bias_relu<<<nd_blocks, T, 0, stream>>>(agg, bs + (size_t)l * D);
        h = agg;   // activated output is next layer's input
    }

    // output head. final h lives in bufs[1] (l=7); bufs[0] and tmp are free.
    float* tmp1 = tmp;
    float* aggs = bufs[0];
    gemv_out<<<(N_NODES * 32 + T - 1) / T, T, 0, stream>>>(h, Wout, tmp1, N_NODES);
    out_init<<<(N_NODES + T - 1) / T, T, 0, stream>>>(tmp1, dinv, aggs, N_NODES);
    out_scatter<<<(n_edges + T - 1) / T, T, 0, stream>>>(src, dst, dinv, tmp1, aggs, n_edges);
    out_finish<<<(N_NODES + T - 1) / T, T, 0, stream>>>(aggs, bout, (float*)d_out, N_NODES);
}